// QuantumConvolutionalModel_81853486727829
// MI455X (gfx1250) — compile-verified
//
#include <hip/hip_runtime.h>
#include <math.h>

// ---------------------------------------------------------------------------
// QCNN statevector simulator for MI455X (gfx1250, wave32).
// State kept as separate re/im f32 planes in d_ws (needs ~8.5 MB):
//   sre[2^20], sim[2^20], U[60*8], G[16*512], thTbl[16*16]
// Strategy: fuse RX*RZ*RX per wire -> 2x2 U; fuse 4 wires -> 16x16 complex
// matrix applied with chained V_WMMA_F32_16X16X4_F32; fuse RZZ diagonal into
// the last group pass of each depth (zero-padded coupling table, branch-free);
// fuse all pool CNOTs into one swap pass; first pass reads inputs directly.
// ---------------------------------------------------------------------------

typedef __attribute__((ext_vector_type(2))) float v2f;
typedef __attribute__((ext_vector_type(8))) float v8f;

#define NSTATE (1u << 20)   // 16 batch * 2^16 amplitudes

__device__ __forceinline__ unsigned spread4(int v, int b0, int b1, int b2, int b3) {
  return ((unsigned)(v & 1) << b0) | ((unsigned)((v >> 1) & 1) << b1) |
         ((unsigned)((v >> 2) & 1) << b2) | ((unsigned)((v >> 3) & 1) << b3);
}

// ---------------- prep 1: fuse RX(t2)*RZ(t1)*RX(t0) per (layer,depth,wire) --
__global__ void qc_prep_fuse(const float* p0, const float* p1, const float* p2,
                             const float* p3, float* __restrict__ U) {
  int t = threadIdx.x;
  if (t >= 60) return;
  int n, d, j;
  const float* p;
  if (t < 32)      { n = 16; int r = t;      d = r >> 4; j = r & 15; p = p0; }
  else if (t < 48) { n = 8;  int r = t - 32; d = r >> 3; j = r & 7;  p = p1; }
  else if (t < 56) { n = 4;  int r = t - 48; d = r >> 2; j = r & 3;  p = p2; }
  else             { n = 2;  int r = t - 56; d = r >> 1; j = r & 1;  p = p3; }
  int base = 4 * j + d * 4 * n;
  float c0 = cosf(0.5f * p[base + 0]), s0 = sinf(0.5f * p[base + 0]);
  float cz = cosf(0.5f * p[base + 1]), sz = sinf(0.5f * p[base + 1]);
  float c2 = cosf(0.5f * p[base + 2]), s2 = sinf(0.5f * p[base + 2]);
  // RX0 = [[c0, -i s0],[-i s0, c0]], A = diag(e^{-i},e^{+i}) * RX0
  float a00r =  cz * c0, a00i = -sz * c0;
  float a01r = -s0 * sz, a01i = -s0 * cz;
  float a10r =  s0 * sz, a10i = -s0 * cz;
  float a11r =  cz * c0, a11i =  sz * c0;
  // U = RX2 * A,  RX2 row mixing with (c2, -i s2)
  float u[8];
  u[0] = c2 * a00r + s2 * a10i;  u[1] = c2 * a00i - s2 * a10r;   // U00
  u[2] = c2 * a01r + s2 * a11i;  u[3] = c2 * a01i - s2 * a11r;   // U01
  u[4] = s2 * a00i + c2 * a10r;  u[5] = -s2 * a00r + c2 * a10i;  // U10
  u[6] = s2 * a01i + c2 * a11r;  u[7] = -s2 * a01r + c2 * a11i;  // U11
  float* o = U + t * 8;
#pragma unroll
  for (int k = 0; k < 8; ++k) o[k] = u[k];
}

// ---------------- prep 2: 16x16 group matrices (tensor product of 4 U's) ----
struct GCfg { int uIdx[16][4]; };  // -1 => identity slot

__global__ void qc_prep_group(const float* __restrict__ U, float* __restrict__ G, GCfg cfg) {
  int g = blockIdx.x;         // 16 group instances
  int t = threadIdx.x;        // 256 = 16x16 elements
  int i = t >> 4, j = t & 15;
  float pr = 1.f, pi = 0.f;
#pragma unroll
  for (int s = 0; s < 4; ++s) {
    int ib = (i >> s) & 1, jb = (j >> s) & 1;
    float fr, fi;
    int u = cfg.uIdx[g][s];
    if (u < 0) { fr = (ib == jb) ? 1.f : 0.f; fi = 0.f; }
    else { const float* up = U + u * 8 + (ib * 2 + jb) * 2; fr = up[0]; fi = up[1]; }
    float nr = pr * fr - pi * fi, ni = pr * fi + pi * fr;
    pr = nr; pi = ni;
  }
  G[g * 512 + i * 16 + j]       = pr;
  G[g * 512 + 256 + i * 16 + j] = pi;
}

// ---------------- prep 3: zero-padded RZZ half-angle table per pass ---------
struct ThCfg { int layer[16]; int nCoup[16]; int cth[16][16]; };

__global__ void qc_prep_theta(const float* p0, const float* p1, const float* p2,
                              const float* p3, float* __restrict__ tbl, ThCfg cfg) {
  int t = threadIdx.x;        // 256 = 16 passes * 16 slots
  int p = t >> 4, c = t & 15;
  const float* th[4] = {p0, p1, p2, p3};
  float v = 0.f;
  if (c < cfg.nCoup[p]) v = 0.5f * th[cfg.layer[p]][cfg.cth[p][c]];
  tbl[t] = v;
}

// ---------------- main pass: 16x16 complex matmul on 4 state bits + diag ----
struct PassCfg {
  int gb0, gb1, gb2, gb3;  // group bit positions (slot order = matrix bit order)
  int pb[16];              // non-group bits ascending: [0..3]=row bits, [4..15]=tile bits
  int gOff;                // float offset of this pass's matrix in G
  int nCoup;               // 0 => no RZZ epilogue (padded slots: th=0, ca=cb=0)
  int ca[16], cb[16], cth[16];
};

__global__ __launch_bounds__(256) void qc_group_pass(const float* __restrict__ sreIn,
                                                     const float* __restrict__ simIn,
                                                     float* __restrict__ sreOut,
                                                     float* __restrict__ simOut,
                                                     const float* __restrict__ G,
                                                     const float* __restrict__ thTbl,
                                                     PassCfg cfg) {
  const int lane = threadIdx.x & 31;
  const int wave = threadIdx.x >> 5;
  const int tile = blockIdx.x * 8 + wave;   // 4096 tiles exactly
  const int m    = lane & 15;               // A row / B col / D col index
  const int half = lane >> 4;
  const int kb   = 2 * half;                // ISA A-layout: lanes16-31 hold K=2,3

  // deposit tile's 12 bits into the 12 high non-group positions (uniform)
  unsigned depH = 0;
#pragma unroll
  for (int i = 0; i < 12; ++i) depH |= ((unsigned)((tile >> i) & 1)) << cfg.pb[4 + i];

  const unsigned rowOff = spread4(m, cfg.pb[0], cfg.pb[1], cfg.pb[2], cfg.pb[3]);
  const unsigned baseIn = depH + rowOff;

  const float* Gre = G + cfg.gOff;
  const float* Gim = Gre + 256;

  v2f aRe[4], aIm[4], bRe[4], bIm[4];
#pragma unroll
  for (int q = 0; q < 4; ++q) {
    const int k0 = 4 * q + kb;
    const unsigned o0 = baseIn + spread4(k0,     cfg.gb0, cfg.gb1, cfg.gb2, cfg.gb3);
    const unsigned o1 = baseIn + spread4(k0 + 1, cfg.gb0, cfg.gb1, cfg.gb2, cfg.gb3);
    aRe[q].x = sreIn[o0]; aRe[q].y = sreIn[o1];
    aIm[q].x = simIn[o0]; aIm[q].y = simIn[o1];
    // B = M^T slice: B[k][n] = M[n][k]; row n contiguous pair (k0,k0+1)
    const float* gr = Gre + m * 16 + k0;
    const float* gi = Gim + m * 16 + k0;
    bRe[q].x = gr[0]; bRe[q].y = gr[1];
    bIm[q].x = gi[0]; bIm[q].y = gi[1];
  }

  v8f cre = {0.f, 0.f, 0.f, 0.f, 0.f, 0.f, 0.f, 0.f};
  v8f cim = {0.f, 0.f, 0.f, 0.f, 0.f, 0.f, 0.f, 0.f};
  // out_im = in_re*Mim^T + in_im*Mre^T ; out_re = in_re*Mre^T + (-in_im)*Mim^T
#pragma unroll
  for (int q = 0; q < 4; ++q) {
    cim = __builtin_amdgcn_wmma_f32_16x16x4_f32(false, aRe[q], false, bIm[q], (short)0, cim, false, false);
    cim = __builtin_amdgcn_wmma_f32_16x16x4_f32(false, aIm[q], false, bRe[q], (short)0, cim, false, false);
    cre = __builtin_amdgcn_wmma_f32_16x16x4_f32(false, aRe[q], false, bRe[q], (short)0, cre, false, false);
  }
#pragma unroll
  for (int q = 0; q < 4; ++q) aIm[q] = -aIm[q];
#pragma unroll
  for (int q = 0; q < 4; ++q)
    cre = __builtin_amdgcn_wmma_f32_16x16x4_f32(false, aIm[q], false, bIm[q], (short)0, cre, false, false);

  // branch-free preload of zero-padded half-angles (uniform scalar loads)
  float th[16];
#pragma unroll
  for (int c = 0; c < 16; ++c) th[c] = thTbl[c];

  const unsigned colOff = spread4(m, cfg.gb0, cfg.gb1, cfg.gb2, cfg.gb3);  // D col = lane&15
#pragma unroll
  for (int v = 0; v < 8; ++v) {
    const int r = v + 8 * half;  // ISA D-layout: VGPR v holds rows v and v+8
    const unsigned addr = depH + spread4(r, cfg.pb[0], cfg.pb[1], cfg.pb[2], cfg.pb[3]) + colOff;
    float re = cre[v], im = cim[v];
    if (cfg.nCoup) {
      float ang = 0.f;
#pragma unroll
      for (int c = 0; c < 16; ++c) {
        int s = ((addr >> cfg.ca[c]) ^ (addr >> cfg.cb[c])) & 1;  // differ -> +1
        ang += s ? th[c] : -th[c];   // padded entries are exactly +/-0
      }
      float sn, cs;
      __sincosf(ang, &sn, &cs);
      float nr = re * cs - im * sn, ni = re * sn + im * cs;
      re = nr; im = ni;
    }
    sreOut[addr] = re;
    simOut[addr] = im;
  }
}

// ---------------- pool: all CNOTs of a layer as one involutive swap ---------
struct PoolCfg { int np; int cbit[8]; int tmask[8]; };

__global__ void qc_pool(float* __restrict__ sre, float* __restrict__ sim, PoolCfg cfg) {
  unsigned i = blockIdx.x * 256u + threadIdx.x;
  unsigned mflip = 0;
#pragma unroll
  for (int p = 0; p < 8; ++p)
    if (p < cfg.np) mflip |= ((i >> cfg.cbit[p]) & 1u) * (unsigned)cfg.tmask[p];
  unsigned j = i ^ mflip;
  if (j > i) {  // one canonical thread per pair performs the swap
    float r0 = sre[i], r1 = sre[j], i0 = sim[i], i1 = sim[j];
    sre[i] = r1; sre[j] = r0; sim[i] = i1; sim[j] = i0;
  }
}

// ---------------- measure <Z> on wire 0 (state bit 15) ----------------------
__global__ void qc_measure(const float* __restrict__ sre, const float* __restrict__ sim,
                           float* __restrict__ out) {
  __shared__ float red[256];
  const int b = blockIdx.x, t = threadIdx.x;
  const float* pr = sre + (unsigned)b * 65536u;
  const float* pi = sim + (unsigned)b * 65536u;
  float acc = 0.f;
  for (int k = t; k < 65536; k += 256) {
    float p = pr[k] * pr[k] + pi[k] * pi[k];
    acc += (k & 32768) ? -p : p;
  }
  red[t] = acc;
  __syncthreads();
  for (int s = 128; s > 0; s >>= 1) {
    if (t < s) red[t] += red[t + s];
    __syncthreads();
  }
  if (t == 0) out[b] = red[0];
}

// ---------------------------------------------------------------------------
extern "C" void kernel_launch(void* const* d_in, const int* in_sizes, int n_in,
                              void* d_out, int out_size, void* d_ws, size_t ws_size,
                              hipStream_t stream) {
  (void)in_sizes; (void)n_in; (void)out_size; (void)ws_size;
  const float* x_re = (const float*)d_in[0];
  const float* x_im = (const float*)d_in[1];
  const float* params[4] = {(const float*)d_in[2], (const float*)d_in[3],
                            (const float*)d_in[4], (const float*)d_in[5]};
  float* sre = (float*)d_ws;
  float* sim = sre + NSTATE;
  float* U   = sim + NSTATE;      // 60 fused 2x2 complex unitaries (480 f32)
  float* G   = U + 480;           // 16 group matrices * (256 re + 256 im)
  float* TH  = G + 16 * 512;      // 16 passes * 16 zero-padded half-angles

  // ---- build the static schedule (deterministic every call) ----
  GCfg gc = {};
  ThCfg tc = {};
  PassCfg passes[16] = {};
  int passStart[4], passEnd[4];
  PoolCfg pools[4] = {};
  const int layerMatBase[4] = {0, 32, 48, 56};

  int act[16]; for (int i = 0; i < 16; ++i) act[i] = i;
  int n = 16, gInst = 0, nPasses = 0;
  for (int layer = 0; layer < 4; ++layer) {
    passStart[layer] = nPasses;
    const int nG = (n + 3) / 4;
    for (int d = 0; d < 2; ++d) {
      for (int g = 0; g < nG; ++g) {
        int gb[4];
        for (int s = 0; s < 4; ++s) {
          int j = g * 4 + s;
          if (j < n) { gb[s] = 15 - act[j]; gc.uIdx[gInst][s] = layerMatBase[layer] + d * n + j; }
          else       { gb[s] = -1;          gc.uIdx[gInst][s] = -1; }
        }
        for (int s = 0; s < 4; ++s)  // pad identity slots with unused low bits
          if (gb[s] < 0)
            for (int b = 0; b < 20; ++b) {
              bool used = false;
              for (int s2 = 0; s2 < 4; ++s2) if (gb[s2] == b) used = true;
              if (!used) { gb[s] = b; break; }
            }
        PassCfg& pc = passes[nPasses];
        pc.gb0 = gb[0]; pc.gb1 = gb[1]; pc.gb2 = gb[2]; pc.gb3 = gb[3];
        int cnt = 0;
        for (int b = 0; b < 20; ++b) {
          bool ing = false;
          for (int s = 0; s < 4; ++s) if (gb[s] == b) ing = true;
          if (!ing) pc.pb[cnt++] = b;
        }
        pc.gOff = gInst * 512;
        pc.nCoup = 0;
        if (g == nG - 1) {  // RZZ diagonal fused into the depth's last pass
          pc.nCoup = n;
          for (int j = 0; j < n; ++j) {
            pc.ca[j]  = 15 - act[j];
            pc.cb[j]  = 15 - act[(j + 1) % n];
            pc.cth[j] = 4 * j + d * 4 * n + 3;
          }
        }
        tc.layer[nPasses] = layer;
        tc.nCoup[nPasses] = pc.nCoup;
        for (int c = 0; c < 16; ++c) tc.cth[nPasses][c] = pc.cth[c];
        ++nPasses; ++gInst;
      }
    }
    passEnd[layer] = nPasses;
    PoolCfg& pl = pools[layer];
    pl.np = n / 2;
    for (int p = 0; p < n / 2; ++p) {
      pl.cbit[p]  = 15 - act[2 * p];
      pl.tmask[p] = 1 << (15 - act[2 * p + 1]);
    }
    int nn = 0;
    for (int j = 0; j < n; j += 2) act[nn++] = act[j];
    n = nn;
  }

  // ---- launch ----
  qc_prep_fuse<<<1, 64, 0, stream>>>(params[0], params[1], params[2], params[3], U);
  qc_prep_group<<<16, 256, 0, stream>>>(U, G, gc);
  qc_prep_theta<<<1, 256, 0, stream>>>(params[0], params[1], params[2], params[3], TH, tc);
  for (int layer = 0; layer < 4; ++layer) {
    for (int p = passStart[layer]; p < passEnd[layer]; ++p) {
      const float* inRe = (p == 0) ? x_re : sre;   // first pass streams inputs directly
      const float* inIm = (p == 0) ? x_im : sim;
      qc_group_pass<<<512, 256, 0, stream>>>(inRe, inIm, sre, sim, G, TH + p * 16, passes[p]);
    }
    qc_pool<<<NSTATE / 256, 256, 0, stream>>>(sre, sim, pools[layer]);
  }
  qc_measure<<<16, 256, 0, stream>>>(sre, sim, (float*)d_out);
}